// RPN_64931315581597
// MI455X (gfx1250) — compile-verified
//
#include <hip/hip_runtime.h>
#include <math.h>

// RPN anchor pipeline, fused single kernel.
// Outputs (flat, fp32): proposals[25200*4] | matched_gt[25200*5] | gt_deltas[25200*4]

#define NP3   19200
#define NP4   4800
#define NP5   1200
#define NANCH 25200
#define NGT   64

#define OUT_PROP  0
#define OUT_MATCH 100800   // 25200*4
#define OUT_DELTA 226800   // 25200*(4+5)

#define SCALE_CLAMP 3.332204510175204f   // log(224/8) computed in double, rounded to f32

__global__ __launch_bounds__(256) void rpn_fused_kernel(
    const float* __restrict__ loc3,
    const float* __restrict__ loc4,
    const float* __restrict__ loc5,
    const float* __restrict__ del3,
    const float* __restrict__ del4,
    const float* __restrict__ del5,
    const float* __restrict__ gt,
    float* __restrict__ out)
{
    __shared__ __align__(16) float sgt[NGT * 5];  // 64 GT boxes x (x0,y0,x1,y1,cls) = 1280 B
    __shared__ float sarea[NGT];                  // precomputed GT areas

    const int tid = threadIdx.x;

    // ---- Stage GT table into LDS via CDNA5 async global->LDS DMA path ----
    // 1280 B = 80 lanes x 16 B. VDST VGPR carries the LDS byte offset (low 32
    // bits of the generic shared pointer), VADDR the 64-bit global address.
    if (tid < (NGT * 5) / 4) {
        const float* gsrc = gt + tid * 4;
        unsigned lds_off = (unsigned)(unsigned long long)(void*)&sgt[tid * 4];
        asm volatile("global_load_async_to_lds_b128 %0, %1, off"
                     :: "v"(lds_off), "v"(gsrc) : "memory");
    }
    asm volatile("s_wait_asynccnt 0x0" ::: "memory");
    __syncthreads();

    if (tid < NGT) {
        float g0 = sgt[5*tid+0], g1 = sgt[5*tid+1];
        float g2 = sgt[5*tid+2], g3 = sgt[5*tid+3];
        sarea[tid] = fabsf((g0 - g2) * (g1 - g3));
    }
    __syncthreads();

    const int a = blockIdx.x * 256 + tid;
    if (a >= NANCH) return;

    // ---- Level / location / aspect decode ----
    const float* locs; const float* dels; int li; float scale;
    if (a < NP3)              { locs = loc3; dels = del3; li = a;             scale = 1.0f; }
    else if (a < NP3 + NP4)   { locs = loc4; dels = del4; li = a - NP3;       scale = 2.0f; }
    else                      { locs = loc5; dels = del5; li = a - NP3 - NP4; scale = 4.0f; }

    const int loc = li / 3;
    const int ar  = li - 3 * loc;
    const float lx = locs[2*loc + 0];
    const float ly = locs[2*loc + 1];

    // half extents: sqrt(area/ar)/2 etc. computed in double, rounded to f32;
    // p4/p5 scale by exact powers of two.
    float hw, hh;
    if (ar == 0)      { hw = 22.627416997969522f; hh = 11.313708498984761f; } // ar=0.5
    else if (ar == 1) { hw = 16.0f;               hh = 16.0f;               } // ar=1.0
    else              { hw = 11.313708498984761f; hh = 22.627416997969522f; } // ar=2.0
    hw *= scale; hh *= scale;

    const float a0 = lx - hw, a1 = ly - hh, a2 = lx + hw, a3 = ly + hh;

    // ---- apply_deltas -> proposals ----
    const float tx = dels[4*li + 0];
    const float ty = dels[4*li + 1];
    const float tw = fminf(dels[4*li + 2], SCALE_CLAMP);
    const float th = fminf(dels[4*li + 3], SCALE_CLAMP);
    const float px = (a0 + a2) * 0.5f, py = (a1 + a3) * 0.5f;
    const float pw = fabsf(a0 - a2),   ph = fabsf(a1 - a3);
    const float bx = px + pw * tx,     by = py + ph * ty;
    const float bw = pw * expf(tw),    bh = ph * expf(th);
    out[OUT_PROP + 4*a + 0] = bx - bw * 0.5f;
    out[OUT_PROP + 4*a + 1] = by - bh * 0.5f;
    out[OUT_PROP + 4*a + 2] = bx + bw * 0.5f;
    out[OUT_PROP + 4*a + 3] = by + bw * 0.5f;  // (sic) reference uses bw here

    // ---- IoU vs 64 GT boxes, first-occurrence argmax ----
    const float area1 = fabsf((a0 - a2) * (a1 - a3));
    float bestq = -1.0f; int bi = 0;
    #pragma unroll 8
    for (int g = 0; g < NGT; ++g) {
        const float g0 = sgt[5*g+0], g1 = sgt[5*g+1];
        const float g2 = sgt[5*g+2], g3 = sgt[5*g+3];
        const float xl = fmaxf(a0, g0), yb = fmaxf(a1, g1);
        const float xr = fminf(a2, g2), yt = fminf(a3, g3);
        const float inter = fabsf((xl - xr) * (yb - yt));
        float iou = inter / (area1 + sarea[g] - inter);
        if ((xl > xr) || (yb > yt)) iou = 0.0f;
        if (iou > bestq) { bestq = iou; bi = g; }
    }

    // ---- matched GT with thresholds ----
    float m0 = sgt[5*bi+0], m1 = sgt[5*bi+1], m2 = sgt[5*bi+2];
    float m3 = sgt[5*bi+3], m4 = sgt[5*bi+4];
    if (bestq <= 0.3f)     { m0 = m1 = m2 = m3 = m4 = -1.0f; }
    else if (bestq < 0.7f) { m0 = m1 = m2 = m3 = m4 = -100000000.0f; }
    out[OUT_MATCH + 5*a + 0] = m0;
    out[OUT_MATCH + 5*a + 1] = m1;
    out[OUT_MATCH + 5*a + 2] = m2;
    out[OUT_MATCH + 5*a + 3] = m3;
    out[OUT_MATCH + 5*a + 4] = m4;

    // ---- gt deltas ----
    const float gbx = (m0 + m2) * 0.5f, gby = (m1 + m3) * 0.5f;
    const float gbw = fabsf(m0 - m2),   gbh = fabsf(m1 - m3);
    float dtx = (gbx - px) / pw;
    float dty = (gby - py) / ph;
    float dtw = logf(fmaxf(gbw, 1e-12f) / pw);
    float dth = logf(fmaxf(gbh, 1e-12f) / ph);
    if (m4 < 0.0f) { dtx = dty = dtw = dth = -100000000.0f; }
    out[OUT_DELTA + 4*a + 0] = dtx;
    out[OUT_DELTA + 4*a + 1] = dty;
    out[OUT_DELTA + 4*a + 2] = dtw;
    out[OUT_DELTA + 4*a + 3] = dth;
}

extern "C" void kernel_launch(void* const* d_in, const int* in_sizes, int n_in,
                              void* d_out, int out_size, void* d_ws, size_t ws_size,
                              hipStream_t stream) {
    (void)in_sizes; (void)n_in; (void)out_size; (void)d_ws; (void)ws_size;
    const float* loc3 = (const float*)d_in[0];
    const float* loc4 = (const float*)d_in[1];
    const float* loc5 = (const float*)d_in[2];
    const float* del3 = (const float*)d_in[3];
    const float* del4 = (const float*)d_in[4];
    const float* del5 = (const float*)d_in[5];
    const float* gt   = (const float*)d_in[6];

    const int threads = 256;
    const int blocks  = (NANCH + threads - 1) / threads;  // 99
    rpn_fused_kernel<<<blocks, threads, 0, stream>>>(
        loc3, loc4, loc5, del3, del4, del5, gt, (float*)d_out);
}